// ClassWiseDiscriminativeLoss_1941325217862
// MI455X (gfx1250) — compile-verified
//
#include <hip/hip_runtime.h>
#include <math.h>

// ---------------------------------------------------------------------------
// ClassWiseDiscriminativeLoss for MI455X (gfx1250, wave32)
//   emb:   [B=2][C=16][V=1048576] f32   labels: [B][V] i32   out: scalar f32
// Pass 1: per-class sums via V_WMMA_F32_16X16X4_F32 (onehot x emb), counts.
// Pass 2: per-voxel hinge^2 distance to own class mean, per-class sums.
// Tail:   tiny stats + final combine kernels.
// ---------------------------------------------------------------------------

typedef __attribute__((ext_vector_type(2))) float v2f;
typedef __attribute__((ext_vector_type(8))) float v8f;

#define NUM_BATCH 2
#define NUM_CH    16
#define NUM_CLS   4
#define VOX       (64LL * 128 * 128)   // 1048576 voxels per batch

// workspace layout in floats
#define WS_SUMS     0     // [B][K][C] = 128 floats
#define WS_COUNTS   128   // [B][K]    = 8 (stored as uint32)
#define WS_VARSUM   136   // [B][K]    = 8 floats
#define WS_MEANS    144   // [B][K][C] = 128 floats
#define WS_NPRES    272   // [B]       = 2 floats
#define WS_DISTREG  274   // [B]       = 2 floats
#define WS_ZERO_N   144   // sums+counts+varsum must be zeroed each call

// pass-1 geometry
#define CHUNK       128                     // voxels per wave iteration
#define WPB1        4                       // waves per block (128 thr)
#define BLK1        256                     // gridDim.x
#define CHUNKS_PB   8192                    // VOX / CHUNK
#define ITERS1      8                       // CHUNKS_PB / (BLK1*WPB1) (exact)
#define VSTRIDE     132                     // LDS row pitch (bank-conflict free)

__global__ void zero_ws_kernel(float* __restrict__ ws) {
    int t = threadIdx.x;
    if (t < WS_ZERO_N) ws[t] = 0.0f;
}

// ---------------------------------------------------------------------------
// Pass 1: sums[b][k][c] = sum_v onehot(lab_v==k) * emb[b][c][v]; counts[b][k]
// WMMA mapping (16x16x4 f32): A = onehot (M=class rows, K=4 voxels),
//                             B = emb tile (K=4 voxels, N=16 channels).
// A layout: lane l holds row M=l&15; VGPR0 = K (l<16 ? 0 : 2), VGPR1 = K+1.
// B layout: lane l holds col N=l&15; VGPR0 = K (l<16 ? 0 : 2), VGPR1 = K+1.
// C/D:      lanes 0-15 rows 0-7 in v[0..7]; class rows 0-3 -> v[0..3].
// ---------------------------------------------------------------------------
__global__ __launch_bounds__(128) void pass1_kernel(
    const float* __restrict__ emb, const int* __restrict__ lab,
    float* __restrict__ sums, unsigned* __restrict__ counts) {
    __shared__ float tile[WPB1][NUM_CH * VSTRIDE];   // [channel][voxel] per wave
    __shared__ int   labs[WPB1][CHUNK];
    __shared__ float bsum[NUM_CLS][NUM_CH];

    const int tid  = threadIdx.x;
    const int lane = tid & 31;
    const int wv   = tid >> 5;
    const int row  = lane & 15;          // class row (A) / channel col (B)
    const int half = lane >> 4;          // selects K pair {0,1} vs {2,3}
    const int b    = blockIdx.y;

    const float* embB = emb + (size_t)b * NUM_CH * VOX;
    const int*   labB = lab + (size_t)b * VOX;

    v8f acc0 = {};
    v8f acc1 = {};
    int c0 = 0, c1 = 0, c2 = 0, c3 = 0;

    const int waveg = blockIdx.x * WPB1 + wv;   // [0, 1024)

    for (int it = 0; it < ITERS1; ++it) {
        const long long vox = (long long)(waveg * ITERS1 + it) * CHUNK;

        // coalesced label load; each lane owns voxels 4*lane..4*lane+3
        int4 l4 = *(const int4*)(labB + vox + 4 * lane);
        *(int4*)&labs[wv][4 * lane] = l4;
        c0 += (l4.x == 0) + (l4.y == 0) + (l4.z == 0) + (l4.w == 0);
        c1 += (l4.x == 1) + (l4.y == 1) + (l4.z == 1) + (l4.w == 1);
        c2 += (l4.x == 2) + (l4.y == 2) + (l4.z == 2) + (l4.w == 2);
        c3 += (l4.x == 3) + (l4.y == 3) + (l4.z == 3) + (l4.w == 3);

        // coalesced 512B-per-channel loads staged into LDS (transpose)
#pragma unroll
        for (int c = 0; c < NUM_CH; ++c) {
            float4 e = *(const float4*)(embB + (size_t)c * VOX + vox + 4 * lane);
            *(float4*)&tile[wv][c * VSTRIDE + 4 * lane] = e;
        }
        __syncthreads();

        // 32 voxel-groups of 4; two accumulators break the WMMA RAW chain
#pragma unroll
        for (int g = 0; g < CHUNK / 4; g += 2) {
            {
                int2 l2 = *(const int2*)&labs[wv][g * 4 + half * 2];
                v2f  bb = *(const v2f*)&tile[wv][row * VSTRIDE + g * 4 + half * 2];
                v2f  aa;
                aa.x = (l2.x == row) ? 1.0f : 0.0f;
                aa.y = (l2.y == row) ? 1.0f : 0.0f;
                acc0 = __builtin_amdgcn_wmma_f32_16x16x4_f32(
                    false, aa, false, bb, (short)0, acc0, false, false);
            }
            {
                int  g1 = g + 1;
                int2 l2 = *(const int2*)&labs[wv][g1 * 4 + half * 2];
                v2f  bb = *(const v2f*)&tile[wv][row * VSTRIDE + g1 * 4 + half * 2];
                v2f  aa;
                aa.x = (l2.x == row) ? 1.0f : 0.0f;
                aa.y = (l2.y == row) ? 1.0f : 0.0f;
                acc1 = __builtin_amdgcn_wmma_f32_16x16x4_f32(
                    false, aa, false, bb, (short)0, acc1, false, false);
            }
        }
        __syncthreads();
    }

    // block-level reduction of per-wave accumulators, then global atomics
    if (tid < NUM_CLS * NUM_CH) bsum[tid >> 4][tid & 15] = 0.0f;
    __syncthreads();
    if (half == 0) {                      // lanes 0-15 hold rows 0-7; rows 0-3 used
#pragma unroll
        for (int r = 0; r < NUM_CLS; ++r)
            atomicAdd(&bsum[r][row], acc0[r] + acc1[r]);
    }
#pragma unroll
    for (int off = 16; off > 0; off >>= 1) {
        c0 += __shfl_xor(c0, off);
        c1 += __shfl_xor(c1, off);
        c2 += __shfl_xor(c2, off);
        c3 += __shfl_xor(c3, off);
    }
    __syncthreads();
    if (tid < NUM_CLS * NUM_CH)
        atomicAdd(&sums[b * 64 + tid], bsum[tid >> 4][tid & 15]);
    if (lane == 0) {
        atomicAdd(&counts[b * 4 + 0], (unsigned)c0);
        atomicAdd(&counts[b * 4 + 1], (unsigned)c1);
        atomicAdd(&counts[b * 4 + 2], (unsigned)c2);
        atomicAdd(&counts[b * 4 + 3], (unsigned)c3);
    }
}

// ---------------------------------------------------------------------------
// Stats: means, present-mask math, pairwise dist + reg terms (tiny)
// ---------------------------------------------------------------------------
__global__ void stats_kernel(float* __restrict__ ws) {
    const unsigned* counts = (const unsigned*)(ws + WS_COUNTS);
    int tid = threadIdx.x;
    if (tid < 128) {
        int b = tid >> 6, k = (tid >> 4) & 3;
        float cnt  = (float)counts[b * 4 + k];
        float safe = fmaxf(cnt, 1.0f);
        ws[WS_MEANS + tid] = ws[WS_SUMS + tid] / safe;
    }
    __syncthreads();
    if (tid < NUM_BATCH) {
        int b = tid;
        const float* m = ws + WS_MEANS + b * 64;
        bool pres[NUM_CLS];
        int  np = 0;
        for (int k = 0; k < NUM_CLS; ++k) {
            pres[k] = (counts[b * 4 + k] > 0u) && (k != 0);
            np += pres[k] ? 1 : 0;
        }
        float n = (float)np;
        float pair_sum = 0.0f;
        for (int i = 0; i < NUM_CLS; ++i)
            for (int j = i + 1; j < NUM_CLS; ++j)
                if (pres[i] && pres[j]) {
                    float dsq = 0.0f;
                    for (int c = 0; c < NUM_CH; ++c) {
                        float d = m[i * 16 + c] - m[j * 16 + c];
                        dsq += d * d;
                    }
                    float hd;
                    if (dsq > 0.0f) {
                        float h = fmaxf(1.5f - sqrtf(dsq), 0.0f);
                        hd = h * h;
                    } else {
                        hd = 1.5f * 1.5f;
                    }
                    pair_sum += hd;
                }
        float dist_loss = (np > 1) ? pair_sum / (n * (n - 1.0f)) : 0.0f;
        float reg = 0.0f;
        for (int k = 0; k < NUM_CLS; ++k)
            if (pres[k]) {
                float sq = 0.0f;
                for (int c = 0; c < NUM_CH; ++c) sq += m[k * 16 + c] * m[k * 16 + c];
                reg += (sq > 0.0f) ? sqrtf(sq) : 0.0f;
            }
        float reg_loss = (np > 0) ? reg / n : 0.0f;
        ws[WS_DISTREG + b] = 1.0f * dist_loss + 0.001f * reg_loss;
        ws[WS_NPRES + b]   = n;
    }
}

// ---------------------------------------------------------------------------
// Pass 2: varsum[b][k] = sum_{v: lab=k} max(||e_v - mu_k|| - 0.5, 0)^2
// Each thread owns 4 consecutive voxels; channel loop is float4-coalesced.
// ---------------------------------------------------------------------------
__global__ __launch_bounds__(256) void pass2_kernel(
    const float* __restrict__ emb, const int* __restrict__ lab,
    const float* __restrict__ means, float* __restrict__ varsum) {
    __shared__ float mu[NUM_CLS * NUM_CH];
    __shared__ float bvs[NUM_CLS];
    const int tid = threadIdx.x;
    const int b   = blockIdx.y;
    if (tid < NUM_CLS * NUM_CH) mu[tid] = means[b * 64 + tid];
    if (tid < NUM_CLS) bvs[tid] = 0.0f;
    __syncthreads();

    const long long t   = (long long)blockIdx.x * 256 + tid;
    const long long vox = 4 * t;
    const float* embB = emb + (size_t)b * NUM_CH * VOX;

    int4 l4 = *(const int4*)(lab + (size_t)b * VOX + vox);
    const int k0 = l4.x & 3, k1 = l4.y & 3, k2 = l4.z & 3, k3 = l4.w & 3;

    float s0 = 0.f, s1 = 0.f, s2 = 0.f, s3 = 0.f;
#pragma unroll
    for (int c = 0; c < NUM_CH; ++c) {
        float4 e = *(const float4*)(embB + (size_t)c * VOX + vox);
        float d0 = e.x - mu[k0 * 16 + c];
        float d1 = e.y - mu[k1 * 16 + c];
        float d2 = e.z - mu[k2 * 16 + c];
        float d3 = e.w - mu[k3 * 16 + c];
        s0 += d0 * d0; s1 += d1 * d1; s2 += d2 * d2; s3 += d3 * d3;
    }
    float h0 = fmaxf(sqrtf(s0) - 0.5f, 0.f); h0 *= h0;
    float h1 = fmaxf(sqrtf(s1) - 0.5f, 0.f); h1 *= h1;
    float h2 = fmaxf(sqrtf(s2) - 0.5f, 0.f); h2 *= h2;
    float h3 = fmaxf(sqrtf(s3) - 0.5f, 0.f); h3 *= h3;

    float vs0 = (k0 == 0) * h0 + (k1 == 0) * h1 + (k2 == 0) * h2 + (k3 == 0) * h3;
    float vs1 = (k0 == 1) * h0 + (k1 == 1) * h1 + (k2 == 1) * h2 + (k3 == 1) * h3;
    float vs2 = (k0 == 2) * h0 + (k1 == 2) * h1 + (k2 == 2) * h2 + (k3 == 2) * h3;
    float vs3 = (k0 == 3) * h0 + (k1 == 3) * h1 + (k2 == 3) * h2 + (k3 == 3) * h3;

#pragma unroll
    for (int off = 16; off > 0; off >>= 1) {
        vs0 += __shfl_xor(vs0, off);
        vs1 += __shfl_xor(vs1, off);
        vs2 += __shfl_xor(vs2, off);
        vs3 += __shfl_xor(vs3, off);
    }
    if ((tid & 31) == 0) {
        atomicAdd(&bvs[0], vs0);
        atomicAdd(&bvs[1], vs1);
        atomicAdd(&bvs[2], vs2);
        atomicAdd(&bvs[3], vs3);
    }
    __syncthreads();
    if (tid < NUM_CLS) atomicAdd(&varsum[b * 4 + tid], bvs[tid]);
}

__global__ void final_kernel(const float* __restrict__ ws, float* __restrict__ out) {
    if (threadIdx.x == 0 && blockIdx.x == 0) {
        const unsigned* counts = (const unsigned*)(ws + WS_COUNTS);
        float total = 0.0f;
        for (int b = 0; b < NUM_BATCH; ++b) {
            float n  = ws[WS_NPRES + b];
            float vl = 0.0f;
            if (n > 0.0f) {
                for (int k = 1; k < NUM_CLS; ++k) {
                    unsigned cnt = counts[b * 4 + k];
                    if (cnt > 0u)
                        vl += ws[WS_VARSUM + b * 4 + k] / fmaxf((float)cnt, 1.0f);
                }
                vl /= n;
            }
            total += vl + ws[WS_DISTREG + b];
        }
        out[0] = total * (1.0f / (float)NUM_BATCH);
    }
}

extern "C" void kernel_launch(void* const* d_in, const int* in_sizes, int n_in,
                              void* d_out, int out_size, void* d_ws, size_t ws_size,
                              hipStream_t stream) {
    const float* emb = (const float*)d_in[0];
    const int*   lab = (const int*)d_in[1];
    float*    ws     = (float*)d_ws;
    float*    sums   = ws + WS_SUMS;
    unsigned* counts = (unsigned*)(ws + WS_COUNTS);
    float*    varsum = ws + WS_VARSUM;
    float*    means  = ws + WS_MEANS;

    zero_ws_kernel<<<1, 256, 0, stream>>>(ws);
    pass1_kernel<<<dim3(BLK1, NUM_BATCH), 128, 0, stream>>>(emb, lab, sums, counts);
    stats_kernel<<<1, 128, 0, stream>>>(ws);
    pass2_kernel<<<dim3(1024, NUM_BATCH), 256, 0, stream>>>(emb, lab, means, varsum);
    final_kernel<<<1, 64, 0, stream>>>(ws, (float*)d_out);
}